// PerformerAttention_49237505081855
// MI455X (gfx1250) — compile-verified
//
#include <hip/hip_runtime.h>
#include <cstddef>

// ---------------------------------------------------------------------------
// Performer (FAVOR+) causal linear attention, f32 end-to-end, CDNA5 WMMA.
// B=2, N=2048, DIM=512, H=8, D=64, M=128
// ---------------------------------------------------------------------------

#define Bb   2
#define Nn   2048
#define DIMd 512
#define Hh   8
#define Dd   64
#define Mm   128
#define PHIR (Bb * Hh * Nn)     // 32768 (per-(b,h,n) rows)

typedef __attribute__((ext_vector_type(2))) float v2f;
typedef __attribute__((ext_vector_type(8))) float v8f;

__device__ __forceinline__ v8f wmma4(v2f a, v2f b, v8f c) {
  // V_WMMA_F32_16X16X4_F32 : D = A(16x4) * B(4x16) + C(16x16), all f32
  return __builtin_amdgcn_wmma_f32_16x16x4_f32(
      false, a, false, b, (short)0, c, false, false);
}
__device__ __forceinline__ v8f zero8() {
  v8f z = {0.f, 0.f, 0.f, 0.f, 0.f, 0.f, 0.f, 0.f};
  return z;
}
__device__ __forceinline__ void cp16(float* dst, const float* src) {
  *reinterpret_cast<float4*>(dst) = *reinterpret_cast<const float4*>(src);
}

#define SCALE_QK   0.35355339059327373f   /* 64^-0.25 */
#define INV_SQRT_M 0.08838834764831845f   /* 1/sqrt(128) */

// ---------------------------------------------------------------------------
// Kernel 1: fused QKV projection.  q/k scaled by D^-0.25, scattered to
// [B,H,N,D] layout.  Tile 128x64, 8 waves, each wave 32x32 (2x2 WMMA frags).
// Grid (32, 24): y-tiles cover cols 0..1535 = [Wq | Wk | Wv].
// ---------------------------------------------------------------------------
__global__ __launch_bounds__(256) void qkv_proj_kernel(
    const float* __restrict__ x,
    const float* __restrict__ Wq, const float* __restrict__ Wk,
    const float* __restrict__ Wv,
    float* __restrict__ qs, float* __restrict__ ks, float* __restrict__ vv) {
  __shared__ float As[128][36];   // row stride 144B (16B aligned)
  __shared__ float Bs[64][36];

  const int tid  = threadIdx.x;
  const int lane = tid & 31, wave = tid >> 5;
  const int half = lane >> 4, l16 = lane & 15;
  const int mbase = blockIdx.x * 128;
  const int cbase = blockIdx.y * 64;          // 0..1535
  const int sel   = cbase >> 9;               // 0=q 1=k 2=v
  const float* W  = (sel == 0) ? Wq : ((sel == 1) ? Wk : Wv);
  const int cloc  = cbase & 511;
  const int wr = wave >> 1;                   // 0..3
  const int wc = wave & 1;                    // 0..1

  v8f acc[2][2];
  acc[0][0] = zero8(); acc[0][1] = zero8();
  acc[1][0] = zero8(); acc[1][1] = zero8();

  for (int k0 = 0; k0 < DIMd; k0 += 32) {
#pragma unroll
    for (int i = 0; i < 4; ++i) {             // 128x32 A tile, float4
      int idx = tid + i * 256, r = idx >> 3, c = (idx & 7) << 2;
      cp16(&As[r][c], &x[(size_t)(mbase + r) * DIMd + k0 + c]);
    }
#pragma unroll
    for (int i = 0; i < 2; ++i) {             // 64x32 B tile (rows = out cols)
      int idx = tid + i * 256, r = idx >> 3, c = (idx & 7) << 2;
      cp16(&Bs[r][c], &W[(size_t)(cloc + r) * DIMd + k0 + c]);
    }
    if (k0 + 32 < DIMd) {                     // prefetch next A tile
      int r = tid >> 1, seg = tid & 1;
      __builtin_prefetch(&x[(size_t)(mbase + r) * DIMd + k0 + 32 + seg * 16],
                         0, 3);
    }
    __syncthreads();
#pragma unroll
    for (int kk = 0; kk < 32; kk += 4) {
      int ak = kk + 2 * half;
      v2f a0, a1, b0, b1;
      a0.x = As[wr * 32 + l16][ak];       a0.y = As[wr * 32 + l16][ak + 1];
      a1.x = As[wr * 32 + 16 + l16][ak];  a1.y = As[wr * 32 + 16 + l16][ak + 1];
      b0.x = Bs[wc * 32 + l16][ak];       b0.y = Bs[wc * 32 + l16][ak + 1];
      b1.x = Bs[wc * 32 + 16 + l16][ak];  b1.y = Bs[wc * 32 + 16 + l16][ak + 1];
      acc[0][0] = wmma4(a0, b0, acc[0][0]);
      acc[0][1] = wmma4(a0, b1, acc[0][1]);
      acc[1][0] = wmma4(a1, b0, acc[1][0]);
      acc[1][1] = wmma4(a1, b1, acc[1][1]);
    }
    __syncthreads();
  }

  float* dst = (sel == 0) ? qs : ((sel == 1) ? ks : vv);
  const float mul = (sel == 2) ? 1.0f : SCALE_QK;
#pragma unroll
  for (int fr = 0; fr < 2; ++fr)
#pragma unroll
    for (int fc = 0; fc < 2; ++fc)
#pragma unroll
      for (int j = 0; j < 8; ++j) {
        int r  = mbase + wr * 32 + fr * 16 + j + 8 * half;
        int cl = cloc + wc * 32 + fc * 16 + l16;
        int h = cl >> 6, d = cl & 63;
        int b = r >> 11, n = r & (Nn - 1);
        dst[(((size_t)(b * Hh + h) * Nn) + n) * Dd + d] = acc[fr][fc][j] * mul;
      }
}

// ---------------------------------------------------------------------------
// Kernel 2: phi features.  proj = Xs @ omega^T ([128 rows] x [K=64] x [M=128]).
// ISQ: subtract per-row max, exp, store phi_q.
// !ISQ: store log-phi, emit per-block max partial (global max done later).
// 8 waves: wave = (rb 0..3) x (cb 0..1); frags 2x4 per wave.
// ---------------------------------------------------------------------------
template <bool ISQ>
__global__ __launch_bounds__(256) void phi_kernel(
    const float* __restrict__ X,      // [PHIR, 64] scaled q or k
    const float* __restrict__ omega,  // [M, D]
    float* __restrict__ out,          // [PHIR, M]
    float* __restrict__ blockmax) {
  __shared__ float Om[Mm][68];        // row stride 272B (16B aligned)
  __shared__ float Xs[128][68];
  __shared__ float ss[128];
  __shared__ float rowm2[128][2];
  __shared__ float wmax[8];

  const int tid = threadIdx.x;
  const int lane = tid & 31, wave = tid >> 5;
  const int half = lane >> 4, l16 = lane & 15;
  const int rowbase = blockIdx.x * 128;
  const int rb = wave >> 1, cb = wave & 1;

#pragma unroll
  for (int i = 0; i < 8; ++i) {       // 128x64 each, float4
    int idx = tid + i * 256, r = idx >> 4, c = (idx & 15) << 2;
    cp16(&Om[r][c], &omega[(size_t)r * Dd + c]);
    cp16(&Xs[r][c], &X[(size_t)(rowbase + r) * Dd + c]);
  }
  __syncthreads();
  if (tid < 128) {
    float s = 0.f;
#pragma unroll
    for (int c = 0; c < Dd; ++c) { float v = Xs[tid][c]; s += v * v; }
    ss[tid] = 0.5f * s;
  }
  __syncthreads();

  v8f acc[2][4];
#pragma unroll
  for (int fr = 0; fr < 2; ++fr)
#pragma unroll
    for (int fc = 0; fc < 4; ++fc) acc[fr][fc] = zero8();

#pragma unroll
  for (int kk = 0; kk < Dd; kk += 4) {
    int ak = kk + 2 * half;
    v2f a0, a1;
    a0.x = Xs[rb * 32 + l16][ak];      a0.y = Xs[rb * 32 + l16][ak + 1];
    a1.x = Xs[rb * 32 + 16 + l16][ak]; a1.y = Xs[rb * 32 + 16 + l16][ak + 1];
    v2f bf[4];
#pragma unroll
    for (int fc = 0; fc < 4; ++fc) {
      bf[fc].x = Om[cb * 64 + fc * 16 + l16][ak];
      bf[fc].y = Om[cb * 64 + fc * 16 + l16][ak + 1];
    }
#pragma unroll
    for (int fc = 0; fc < 4; ++fc) {
      acc[0][fc] = wmma4(a0, bf[fc], acc[0][fc]);
      acc[1][fc] = wmma4(a1, bf[fc], acc[1][fc]);
    }
  }

  if constexpr (ISQ) {
#pragma unroll
    for (int fr = 0; fr < 2; ++fr) {
      float pm[8];
#pragma unroll
      for (int j = 0; j < 8; ++j) pm[j] = -3.4e38f;
#pragma unroll
      for (int fc = 0; fc < 4; ++fc)
#pragma unroll
        for (int j = 0; j < 8; ++j) pm[j] = fmaxf(pm[j], acc[fr][fc][j]);
#pragma unroll
      for (int j = 0; j < 8; ++j) {
        for (int m = 8; m >= 1; m >>= 1)
          pm[j] = fmaxf(pm[j], __shfl_xor(pm[j], m, 32));
        if (l16 == 0) {
          int rl = rb * 32 + fr * 16 + j + 8 * half;
          rowm2[rl][cb] = pm[j] - ss[rl];
        }
      }
    }
    __syncthreads();
#pragma unroll
    for (int fr = 0; fr < 2; ++fr)
#pragma unroll
      for (int fc = 0; fc < 4; ++fc)
#pragma unroll
        for (int j = 0; j < 8; ++j) {
          int rl = rb * 32 + fr * 16 + j + 8 * half;
          int cc = cb * 64 + fc * 16 + l16;
          float mx = fmaxf(rowm2[rl][0], rowm2[rl][1]);
          float lp = acc[fr][fc][j] - ss[rl] - mx;
          out[(size_t)(rowbase + rl) * Mm + cc] =
              __expf(lp) * INV_SQRT_M + 1e-4f;
        }
  } else {
    float pm = -3.4e38f;
#pragma unroll
    for (int fr = 0; fr < 2; ++fr)
#pragma unroll
      for (int fc = 0; fc < 4; ++fc)
#pragma unroll
        for (int j = 0; j < 8; ++j) {
          int rl = rb * 32 + fr * 16 + j + 8 * half;
          int cc = cb * 64 + fc * 16 + l16;
          float lp = acc[fr][fc][j] - ss[rl];
          out[(size_t)(rowbase + rl) * Mm + cc] = lp;
          pm = fmaxf(pm, lp);
        }
    for (int m = 16; m >= 1; m >>= 1) pm = fmaxf(pm, __shfl_xor(pm, m, 32));
    if (lane == 0) wmax[wave] = pm;
    __syncthreads();
    if (tid == 0) {
      float g = wmax[0];
#pragma unroll
      for (int w = 1; w < 8; ++w) g = fmaxf(g, wmax[w]);
      blockmax[blockIdx.x] = g;
    }
  }
}

// ---------------------------------------------------------------------------
// Kernel 3: reduce 256 block maxima -> scalar global max.
// ---------------------------------------------------------------------------
__global__ void reduce_max_kernel(const float* __restrict__ bm,
                                  float* __restrict__ gmax) {
  __shared__ float s[256];
  s[threadIdx.x] = bm[threadIdx.x];
  __syncthreads();
  for (int off = 128; off > 0; off >>= 1) {
    if ((int)threadIdx.x < off)
      s[threadIdx.x] = fmaxf(s[threadIdx.x], s[threadIdx.x + off]);
    __syncthreads();
  }
  if (threadIdx.x == 0) gmax[0] = s[0];
}

// Kernel 4: phi_k = exp(logphi_k - gmax)/sqrt(M) + 1e-4   (in place)
__global__ void exp_k_kernel(float* __restrict__ lp,
                             const float* __restrict__ gmax, int n) {
  int i = blockIdx.x * blockDim.x + threadIdx.x;
  if (i < n) lp[i] = __expf(lp[i] - gmax[0]) * INV_SQRT_M + 1e-4f;
}

// ---------------------------------------------------------------------------
// Kernel 5: chunked causal linear-attention scan.  One block per (b,h).
// Chunk C=64.  State S[M,D] + sk[M] live in LDS across all 32 chunks.
//   A    = mask(Qc Kc^T)            [64x64]   (K=128)
//   O    = A Vc + Qc S_prev         [64x64]   (K=64 then K=128)
//   den  = Qc.sk_prev + rowsum(A) + 1e-6
//   S   += Kc^T Vc ; sk += colsum(Kc)
//   y    = O / den   ->  [B, N, H*D]
// ---------------------------------------------------------------------------
__global__ __launch_bounds__(256) void scan_kernel(
    const float* __restrict__ phi_q, const float* __restrict__ phi_k,
    const float* __restrict__ vv, float* __restrict__ y) {
  __shared__ float S[Mm][66];
  __shared__ float sk[Mm];
  __shared__ float Qc[64][132];       // row stride 528B (16B aligned)
  __shared__ float Kc[64][132];
  __shared__ float Vc[64][68];        // row stride 272B (16B aligned)
  __shared__ float Am[64][66];
  __shared__ float den[64];

  const int tid = threadIdx.x;
  const int lane = tid & 31, wave = tid >> 5;
  const int half = lane >> 4, l16 = lane & 15;
  const int bh = blockIdx.x;            // 0..15
  const int b = bh >> 3, h = bh & 7;

  for (int i = tid; i < Mm * Dd; i += 256) S[i >> 6][i & 63] = 0.f;
  if (tid < Mm) sk[tid] = 0.f;
  __syncthreads();

  const int fr2 = wave >> 1;            // 0..3  (16-row block)
  const int cp  = (wave & 1) * 2;       // col frags {cp, cp+1}

  for (int t = 0; t < Nn / 64; ++t) {
    const int n0 = t * 64;
    const size_t rowoff = ((size_t)bh * Nn + n0);
#pragma unroll
    for (int i = 0; i < 8; ++i) {       // Qc, Kc: 64x128 each, float4
      int idx = tid + i * 256, r = idx >> 5, c = (idx & 31) << 2;
      cp16(&Qc[r][c], &phi_q[(rowoff + r) * Mm + c]);
      cp16(&Kc[r][c], &phi_k[(rowoff + r) * Mm + c]);
    }
#pragma unroll
    for (int i = 0; i < 4; ++i) {       // Vc: 64x64, float4
      int idx = tid + i * 256, r = idx >> 4, c = (idx & 15) << 2;
      cp16(&Vc[r][c], &vv[(rowoff + r) * Dd + c]);
    }
    if (t + 1 < Nn / 64) {              // prefetch next chunk
      const size_t ro2 = rowoff + 64;
      int r = tid >> 2, seg = tid & 3;
      __builtin_prefetch(&phi_q[(ro2 + r) * Mm + seg * 32], 0, 3);
      __builtin_prefetch(&phi_k[(ro2 + r) * Mm + seg * 32], 0, 3);
      if (seg < 2) __builtin_prefetch(&vv[(ro2 + r) * Dd + seg * 32], 0, 3);
    }
    __syncthreads();

    // ---- A = Qc Kc^T (K=128), causal mask, into Am ----
    v8f a2[2];
    a2[0] = zero8(); a2[1] = zero8();
#pragma unroll
    for (int kk = 0; kk < Mm; kk += 4) {
      int ak = kk + 2 * half;
      v2f aq, b0, b1;
      aq.x = Qc[fr2 * 16 + l16][ak];        aq.y = Qc[fr2 * 16 + l16][ak + 1];
      b0.x = Kc[cp * 16 + l16][ak];         b0.y = Kc[cp * 16 + l16][ak + 1];
      b1.x = Kc[(cp + 1) * 16 + l16][ak];   b1.y = Kc[(cp + 1) * 16 + l16][ak + 1];
      a2[0] = wmma4(aq, b0, a2[0]);
      a2[1] = wmma4(aq, b1, a2[1]);
    }
#pragma unroll
    for (int f = 0; f < 2; ++f)
#pragma unroll
      for (int j = 0; j < 8; ++j) {
        int rn = fr2 * 16 + j + 8 * half;
        int cn = (cp + f) * 16 + l16;
        Am[rn][cn] = (cn <= rn) ? a2[f][j] : 0.f;
      }
    __syncthreads();

    // ---- den = Qc.sk + rowsum(Am) + 1e-6 (uses sk PRE-update) ----
    if (tid < 64) {
      float rsum = 0.f;
#pragma unroll
      for (int j = 0; j < 64; ++j) rsum += Am[tid][j];
      float dsum = 0.f;
#pragma unroll
      for (int m = 0; m < Mm; ++m) dsum += Qc[tid][m] * sk[m];
      den[tid] = dsum + rsum + 1e-6f;
    }
    __syncthreads();

    // ---- O = Am Vc + Qc S  (same 64x64 frag layout as A) ----
    v8f o2[2];
    o2[0] = zero8(); o2[1] = zero8();
#pragma unroll
    for (int kk = 0; kk < 64; kk += 4) {  // Am @ Vc
      int ak = kk + 2 * half;
      v2f aa, b0, b1;
      aa.x = Am[fr2 * 16 + l16][ak];      aa.y = Am[fr2 * 16 + l16][ak + 1];
      b0.x = Vc[ak][cp * 16 + l16];       b0.y = Vc[ak + 1][cp * 16 + l16];
      b1.x = Vc[ak][(cp + 1) * 16 + l16]; b1.y = Vc[ak + 1][(cp + 1) * 16 + l16];
      o2[0] = wmma4(aa, b0, o2[0]);
      o2[1] = wmma4(aa, b1, o2[1]);
    }
#pragma unroll
    for (int kk = 0; kk < Mm; kk += 4) {  // Qc @ S_prev
      int ak = kk + 2 * half;
      v2f aq, b0, b1;
      aq.x = Qc[fr2 * 16 + l16][ak];      aq.y = Qc[fr2 * 16 + l16][ak + 1];
      b0.x = S[ak][cp * 16 + l16];        b0.y = S[ak + 1][cp * 16 + l16];
      b1.x = S[ak][(cp + 1) * 16 + l16];  b1.y = S[ak + 1][(cp + 1) * 16 + l16];
      o2[0] = wmma4(aq, b0, o2[0]);
      o2[1] = wmma4(aq, b1, o2[1]);
    }
#pragma unroll
    for (int f = 0; f < 2; ++f)
#pragma unroll
      for (int j = 0; j < 8; ++j) {
        int rn = fr2 * 16 + j + 8 * half;
        int cn = (cp + f) * 16 + l16;
        y[((size_t)b * Nn + (n0 + rn)) * (Hh * Dd) + h * Dd + cn] =
            o2[f][j] / den[rn];
      }
    __syncthreads();   // all S reads done before state update

    // ---- S += Kc^T Vc  (wave w owns M-rows [w*16, w*16+16), 4 col frags) ----
    v8f sa[4];
#pragma unroll
    for (int fc = 0; fc < 4; ++fc)
#pragma unroll
      for (int j = 0; j < 8; ++j)
        sa[fc][j] = S[wave * 16 + j + 8 * half][fc * 16 + l16];
#pragma unroll
    for (int kk = 0; kk < 64; kk += 4) {
      int ak = kk + 2 * half;
      v2f at;  // A = Kc^T : A[m][k] = Kc[k][m]
      at.x = Kc[ak][wave * 16 + l16];
      at.y = Kc[ak + 1][wave * 16 + l16];
      v2f bv[4];
#pragma unroll
      for (int fc = 0; fc < 4; ++fc) {
        bv[fc].x = Vc[ak][fc * 16 + l16];
        bv[fc].y = Vc[ak + 1][fc * 16 + l16];
      }
#pragma unroll
      for (int fc = 0; fc < 4; ++fc) sa[fc] = wmma4(at, bv[fc], sa[fc]);
    }
#pragma unroll
    for (int fc = 0; fc < 4; ++fc)
#pragma unroll
      for (int j = 0; j < 8; ++j)
        S[wave * 16 + j + 8 * half][fc * 16 + l16] = sa[fc][j];

    if (tid < Mm) {                      // sk += colsum(Kc)
      float s = 0.f;
#pragma unroll
      for (int j = 0; j < 64; ++j) s += Kc[j][tid];
      sk[tid] += s;
    }
    __syncthreads();
  }
}

// ---------------------------------------------------------------------------
// Kernel 6: out = y @ Wo^T + bo.  Tile 128x64, grid (32, 8).
// ---------------------------------------------------------------------------
__global__ __launch_bounds__(256) void out_proj_kernel(
    const float* __restrict__ yin, const float* __restrict__ Wo,
    const float* __restrict__ bo, float* __restrict__ out) {
  __shared__ float As[128][36];
  __shared__ float Bs[64][36];

  const int tid = threadIdx.x;
  const int lane = tid & 31, wave = tid >> 5;
  const int half = lane >> 4, l16 = lane & 15;
  const int mbase = blockIdx.x * 128;
  const int cbase = blockIdx.y * 64;
  const int wr = wave >> 1, wc = wave & 1;

  v8f acc[2][2];
  acc[0][0] = zero8(); acc[0][1] = zero8();
  acc[1][0] = zero8(); acc[1][1] = zero8();

  for (int k0 = 0; k0 < DIMd; k0 += 32) {
#pragma unroll
    for (int i = 0; i < 4; ++i) {
      int idx = tid + i * 256, r = idx >> 3, c = (idx & 7) << 2;
      cp16(&As[r][c], &yin[(size_t)(mbase + r) * DIMd + k0 + c]);
    }
#pragma unroll
    for (int i = 0; i < 2; ++i) {
      int idx = tid + i * 256, r = idx >> 3, c = (idx & 7) << 2;
      cp16(&Bs[r][c], &Wo[(size_t)(cbase + r) * DIMd + k0 + c]);
    }
    if (k0 + 32 < DIMd) {
      int r = tid >> 1, seg = tid & 1;
      __builtin_prefetch(&yin[(size_t)(mbase + r) * DIMd + k0 + 32 + seg * 16],
                         0, 3);
    }
    __syncthreads();
#pragma unroll
    for (int kk = 0; kk < 32; kk += 4) {
      int ak = kk + 2 * half;
      v2f a0, a1, b0, b1;
      a0.x = As[wr * 32 + l16][ak];       a0.y = As[wr * 32 + l16][ak + 1];
      a1.x = As[wr * 32 + 16 + l16][ak];  a1.y = As[wr * 32 + 16 + l16][ak + 1];
      b0.x = Bs[wc * 32 + l16][ak];       b0.y = Bs[wc * 32 + l16][ak + 1];
      b1.x = Bs[wc * 32 + 16 + l16][ak];  b1.y = Bs[wc * 32 + 16 + l16][ak + 1];
      acc[0][0] = wmma4(a0, b0, acc[0][0]);
      acc[0][1] = wmma4(a0, b1, acc[0][1]);
      acc[1][0] = wmma4(a1, b0, acc[1][0]);
      acc[1][1] = wmma4(a1, b1, acc[1][1]);
    }
    __syncthreads();
  }
#pragma unroll
  for (int fr = 0; fr < 2; ++fr)
#pragma unroll
    for (int fc = 0; fc < 2; ++fc)
#pragma unroll
      for (int j = 0; j < 8; ++j) {
        int r = mbase + wr * 32 + fr * 16 + j + 8 * half;
        int c = cbase + wc * 32 + fc * 16 + l16;
        out[(size_t)r * DIMd + c] = acc[fr][fc][j] + bo[c];
      }
}

// ---------------------------------------------------------------------------
extern "C" void kernel_launch(void* const* d_in, const int* in_sizes, int n_in,
                              void* d_out, int out_size, void* d_ws,
                              size_t ws_size, hipStream_t stream) {
  (void)in_sizes; (void)n_in; (void)out_size; (void)ws_size;
  const float* x     = (const float*)d_in[0];
  const float* omega = (const float*)d_in[1];
  const float* Wq    = (const float*)d_in[2];
  const float* Wk    = (const float*)d_in[3];
  const float* Wv    = (const float*)d_in[4];
  const float* Wo    = (const float*)d_in[5];
  const float* bo    = (const float*)d_in[6];
  float* out = (float*)d_out;

  float* ws = (float*)d_ws;
  size_t o = 0;
  float* qs   = ws + o; o += (size_t)Bb * Hh * Nn * Dd;   // scaled q
  float* ks   = ws + o; o += (size_t)Bb * Hh * Nn * Dd;   // scaled k
  float* vv   = ws + o; o += (size_t)Bb * Hh * Nn * Dd;   // v
  float* pq   = ws + o; o += (size_t)PHIR * Mm;           // phi_q
  float* pk   = ws + o; o += (size_t)PHIR * Mm;           // log-phi_k -> phi_k
  float* yy   = ws + o; o += (size_t)Bb * Nn * Hh * Dd;   // attention out
  float* bmax = ws + o; o += 256;
  float* gmax = ws + o; o += 1;

  qkv_proj_kernel<<<dim3(32, 24), 256, 0, stream>>>(x, Wq, Wk, Wv, qs, ks, vv);
  phi_kernel<true><<<PHIR / 128, 256, 0, stream>>>(qs, omega, pq, nullptr);
  phi_kernel<false><<<PHIR / 128, 256, 0, stream>>>(ks, omega, pk, bmax);
  reduce_max_kernel<<<1, 256, 0, stream>>>(bmax, gmax);
  const int nk = PHIR * Mm;
  exp_k_kernel<<<(nk + 255) / 256, 256, 0, stream>>>(pk, gmax, nk);
  scan_kernel<<<Bb * Hh, 256, 0, stream>>>(pq, pk, vv, yy);
  out_proj_kernel<<<dim3(32, 8), 256, 0, stream>>>(yy, Wo, bo, out);
}